// GNNLayer1_86526411145928
// MI455X (gfx1250) — compile-verified
//
#include <hip/hip_runtime.h>
#include <math.h>

#define NN   50000
#define EE   800000
#define ET   (EE + NN)     // edges incl. self loops
#define INC  128
#define CC   64
#define NEGS 0.2f

typedef float v2f __attribute__((ext_vector_type(2)));
typedef float v8f __attribute__((ext_vector_type(8)));

// ---------------- monotonic float<->uint encoding for atomic max ------------
__device__ __forceinline__ unsigned fenc(float f) {
    unsigned u = __float_as_uint(f);
    return (u >> 31) ? ~u : (u | 0x80000000u);
}
__device__ __forceinline__ float fdec(unsigned u) {
    return __uint_as_float((u >> 31) ? (u & 0x7FFFFFFFu) : ~u);
}

// ---------------- 0: init out / denom / amax --------------------------------
__global__ __launch_bounds__(256)
void init_ws_kernel(float* __restrict__ out, float* __restrict__ denom,
                    unsigned* __restrict__ amax) {
    int i = blockIdx.x * blockDim.x + threadIdx.x;
    if (i < NN * CC) out[i] = 0.0f;
    if (i < NN) { denom[i] = 0.0f; amax[i] = 0x007FFFFFu; /* fenc(-inf) */ }
}

// ---------------- 1: g = x @ W via V_WMMA_F32_16X16X4_F32 -------------------
// One wave computes a 16x64 slab of g. 50000 rows = 3125 * 16 exactly.
__global__ __launch_bounds__(32)
void gemm_wmma_kernel(const float* __restrict__ x, const float* __restrict__ W,
                      float* __restrict__ g) {
    const int rb   = blockIdx.x;        // row tile, 16 rows
    const int lane = threadIdx.x;       // 0..31
    const int half = lane >> 4;         // 0: K={0,1}, 1: K={2,3}
    const int l15  = lane & 15;

    v8f acc0 = {}; v8f acc1 = {}; v8f acc2 = {}; v8f acc3 = {};

    const float* xrow = x + (rb * 16 + l15) * INC + half * 2;  // A: M=l15
    for (int k0 = 0; k0 < INC; k0 += 4) {
        // A fragment: 16x4 fp32 (lanes 0-15: K=k0,k0+1; lanes 16-31: K=k0+2,k0+3)
        v2f a; a.x = xrow[k0]; a.y = xrow[k0 + 1];
        // B fragments (4x16 fp32) for 4 column tiles, W is [128 x 64] row-major
        const float* wp = W + (k0 + half * 2) * CC + l15;
        v2f b0, b1, b2, b3;
        b0.x = wp[0];  b0.y = wp[0  + CC];
        b1.x = wp[16]; b1.y = wp[16 + CC];
        b2.x = wp[32]; b2.y = wp[32 + CC];
        b3.x = wp[48]; b3.y = wp[48 + CC];
        acc0 = __builtin_amdgcn_wmma_f32_16x16x4_f32(false, a, false, b0, (short)0, acc0, false, false);
        acc1 = __builtin_amdgcn_wmma_f32_16x16x4_f32(false, a, false, b1, (short)0, acc1, false, false);
        acc2 = __builtin_amdgcn_wmma_f32_16x16x4_f32(false, a, false, b2, (short)0, acc2, false, false);
        acc3 = __builtin_amdgcn_wmma_f32_16x16x4_f32(false, a, false, b3, (short)0, acc3, false, false);
    }
    // D layout: VGPR v -> M = v + half*8, N = l15
    #pragma unroll
    for (int v = 0; v < 8; ++v) {
        float* op = g + (rb * 16 + v + half * 8) * CC + l15;
        op[0]  = acc0[v];
        op[16] = acc1[v];
        op[32] = acc2[v];
        op[48] = acc3[v];
    }
}

// ---------------- 2: per-node scalars s_i, s_j ------------------------------
__global__ __launch_bounds__(256)
void node_scores_kernel(const float* __restrict__ g, const float* __restrict__ corrs,
                        const float* __restrict__ att_i, const float* __restrict__ att_j,
                        const float* __restrict__ aem_i, const float* __restrict__ aem_j,
                        float* __restrict__ s_i, float* __restrict__ s_j) {
    int node = (blockIdx.x * blockDim.x + threadIdx.x) >> 5;
    int lane = threadIdx.x & 31;
    if (node >= NN) return;
    const float* gr = g + node * CC;
    const float* cr = corrs + node * CC;   // K == 64 == CC
    float si = 0.0f, sj = 0.0f;
    #pragma unroll
    for (int c = lane; c < CC; c += 32) {
        float gv = gr[c], cv = cr[c];
        si += gv * att_i[c] + cv * aem_i[c];
        sj += gv * att_j[c] + cv * aem_j[c];
    }
    #pragma unroll
    for (int off = 16; off > 0; off >>= 1) {
        si += __shfl_xor(si, off, 32);
        sj += __shfl_xor(sj, off, 32);
    }
    if (lane == 0) { s_i[node] = si; s_j[node] = sj; }
}

// ---------------- 3: alpha = leaky_relu(s_i[dst]+s_j[src]); seg max ---------
__global__ __launch_bounds__(256)
void edge_alpha_kernel(const int* __restrict__ ei, const float* __restrict__ s_i,
                       const float* __restrict__ s_j, float* __restrict__ alpha,
                       unsigned* __restrict__ amax) {
    int e = blockIdx.x * blockDim.x + threadIdx.x;
    if (e >= ET) return;
    int src, dst;
    if (e < EE) { src = ei[e]; dst = ei[EE + e]; } else { src = dst = e - EE; }
    float a = s_i[dst] + s_j[src];
    a = (a > 0.0f) ? a : NEGS * a;
    alpha[e] = a;
    atomicMax(amax + dst, fenc(a));
}

// ---------------- 4: ex = exp(alpha - max[dst]); seg sum --------------------
__global__ __launch_bounds__(256)
void edge_exp_kernel(const int* __restrict__ ei, const unsigned* __restrict__ amax,
                     float* __restrict__ alpha_ex, float* __restrict__ denom) {
    int e = blockIdx.x * blockDim.x + threadIdx.x;
    if (e >= ET) return;
    int dst = (e < EE) ? ei[EE + e] : (e - EE);
    float ex = expf(alpha_ex[e] - fdec(amax[dst]));
    alpha_ex[e] = ex;
    atomicAdd(denom + dst, ex);
}

// ---------------- 5: scatter-add weighted messages --------------------------
// 16 lanes per edge, float4 per lane (64 channels)
__global__ __launch_bounds__(256)
void message_kernel(const int* __restrict__ ei, const float* __restrict__ g,
                    const float* __restrict__ ex, const float* __restrict__ denom,
                    float* __restrict__ out) {
    int tid = blockIdx.x * blockDim.x + threadIdx.x;
    int e = tid >> 4;
    int q = tid & 15;
    if (e >= ET) return;
    int src, dst;
    if (e < EE) { src = ei[e]; dst = ei[EE + e]; } else { src = dst = e - EE; }
    float coef = ex[e] / (denom[dst] + 1e-16f);
    float4 gv = ((const float4*)(g + src * CC))[q];
    float* op = out + dst * CC + q * 4;
    atomicAdd(op + 0, gv.x * coef);
    atomicAdd(op + 1, gv.y * coef);
    atomicAdd(op + 2, gv.z * coef);
    atomicAdd(op + 3, gv.w * coef);
}

// ---------------- 6: out = relu(out + bias) ---------------------------------
__global__ __launch_bounds__(256)
void finalize_kernel(float* __restrict__ out, const float* __restrict__ bias) {
    int i = blockIdx.x * blockDim.x + threadIdx.x;
    if (i >= NN * CC) return;
    float v = out[i] + bias[i & (CC - 1)];
    out[i] = (v > 0.0f) ? v : 0.0f;
}

extern "C" void kernel_launch(void* const* d_in, const int* in_sizes, int n_in,
                              void* d_out, int out_size, void* d_ws, size_t ws_size,
                              hipStream_t stream) {
    (void)in_sizes; (void)n_in; (void)out_size; (void)ws_size;
    const float* x      = (const float*)d_in[0];   // [N,128]
    const int*   ei     = (const int*)  d_in[1];   // [2,E]
    const float* corrs  = (const float*)d_in[2];   // [N,64]
    const float* W      = (const float*)d_in[3];   // [128,64]
    const float* att_i  = (const float*)d_in[4];   // [64]
    const float* att_j  = (const float*)d_in[5];   // [64]
    const float* aem_i  = (const float*)d_in[6];   // [64]
    const float* aem_j  = (const float*)d_in[7];   // [64]
    const float* bias   = (const float*)d_in[8];   // [64]
    float* out = (float*)d_out;                    // [N,64]

    // workspace layout (floats)
    float*    ws    = (float*)d_ws;
    float*    g     = ws;                    // N*64
    float*    s_i   = g + (size_t)NN * CC;   // N
    float*    s_j   = s_i + NN;              // N
    unsigned* amax  = (unsigned*)(s_j + NN); // N
    float*    denom = (float*)(amax + NN);   // N
    float*    aex   = denom + NN;            // E+N (alpha, then exp in place)

    init_ws_kernel<<<(NN * CC + 255) / 256, 256, 0, stream>>>(out, denom, amax);
    gemm_wmma_kernel<<<NN / 16, 32, 0, stream>>>(x, W, g);
    node_scores_kernel<<<(NN * 32 + 255) / 256, 256, 0, stream>>>(
        g, corrs, att_i, att_j, aem_i, aem_j, s_i, s_j);
    edge_alpha_kernel<<<(ET + 255) / 256, 256, 0, stream>>>(ei, s_i, s_j, aex, amax);
    edge_exp_kernel<<<(ET + 255) / 256, 256, 0, stream>>>(ei, amax, aex, denom);
    message_kernel<<<(ET * 16 + 255) / 256, 256, 0, stream>>>(ei, g, aex, denom, out);
    finalize_kernel<<<(NN * CC + 255) / 256, 256, 0, stream>>>(out, bias);
}